// GCM_V0_80788334838520
// MI455X (gfx1250) — compile-verified
//
#include <hip/hip_runtime.h>

// Round 3: separable Gaussian pyramid on gfx1250.
//  - both 1-D passes on the fp32 matrix pipe (v_wmma_f32_16x16x4_f32)
//  - compile-time specialization on halo radius p: branch-free WMMA chains
//  - interior tiles staged via global_load_async_to_lds_b64 (ASYNCcnt path)
//  - zero-padded tap table -> branchless banded coefficients, shared by passes

typedef float v2f __attribute__((ext_vector_type(2)));
typedef float v8f __attribute__((ext_vector_type(8)));

#define NK    21
#define MAXK  21
#define BATCH 8
#define CHN   3
#define HDIM  512
#define WDIM  512
#define OCH   66           // 3 identity + 21*3 blurred channels
#define TIW   36           // staged tile edge: 16 + 2*pmax, pmax = 10

__device__ __forceinline__ int refl(int v, int n) {
    // jnp.pad mode="reflect": -1 -> 1, n -> n-2 (single reflection, p <= 10)
    v = (v < 0) ? -v : v;
    v = (v >= n) ? (2 * n - 2 - v) : v;
    return v;
}

__device__ __forceinline__ int ksize_of(int i) {
    // Python: int(3 + (i*(18/19)) // 2 * 2), cap at 21 never binds
    double t = (double)i * (18.0 / 19.0);
    int h = (int)(t * 0.5);            // floor, t >= 0
    int k = 3 + 2 * h;
    return (k > MAXK) ? MAXK : k;
}

// ---- identity channels: out[b, 0:3, :, :] = x ------------------------------
__global__ void __launch_bounds__(256)
identity_copy(const float4* __restrict__ x, float4* __restrict__ out) {
    size_t idx = (size_t)blockIdx.x * blockDim.x + threadIdx.x;
    const size_t plane4 = (size_t)HDIM * WDIM / 4;
    size_t plane = idx / plane4;
    if (plane >= (size_t)BATCH * CHN) return;
    size_t rem = idx - plane * plane4;
    size_t b  = plane / CHN;
    size_t ch = plane - b * CHN;
    out[(b * OCH + ch) * plane4 + rem] = x[idx];
}

// ---- both separable passes, fully specialized on halo radius P -------------
template <int P>
__device__ __forceinline__ void blur_passes(
    const float* __restrict__ s_x, float* __restrict__ s_tmp,
    const float* __restrict__ s_gpad, float* __restrict__ op,
    int half, int m)
{
    constexpr int R   = 16 + 2 * P;    // K extent of both passes
    constexpr int CHK = (R + 3) / 4;   // K chunks of 4
    constexpr int NTT = (R + 15) / 16; // tmp row tiles

    const int q0base = 2 * half;       // lane holds K pair (4t+q0base, +1)

    // banded coefficients, shared by both passes: Wh[q][n] = Wv[n][q] = g[q-n]
    v2f wc[CHK];
    const int dbase = 16 + q0base - m;
    #pragma unroll
    for (int t = 0; t < CHK; ++t) {
        wc[t].x = s_gpad[dbase + 4 * t];
        wc[t].y = s_gpad[dbase + 4 * t + 1];
    }

    // ---- pass 1 (horizontal): tmp(R x 16) = X * Wh -------------------------
    #pragma unroll
    for (int tt = 0; tt < NTT; ++tt) {
        int tr = 10 - P + 16 * tt + m;
        if (tt == NTT - 1) tr = min(tr, TIW - 1);  // garbage rows: clamp in-tile
        const int abase = tr * TIW + (10 - P) + q0base;
        v2f a[CHK];
        #pragma unroll
        for (int t = 0; t < CHK; ++t) {            // one load clause, imm offsets
            a[t].x = s_x[abase + 4 * t];
            a[t].y = s_x[abase + 4 * t + 1];
        }
        v8f acc = {0.f, 0.f, 0.f, 0.f, 0.f, 0.f, 0.f, 0.f};
        #pragma unroll
        for (int t = 0; t < CHK; ++t)              // back-to-back WMMA chain
            acc = __builtin_amdgcn_wmma_f32_16x16x4_f32(
                false, a[t], false, wc[t], (short)0, acc, false, false);
        const int sb = (tt * 16 + half * 8) * 16 + m;
        #pragma unroll
        for (int j = 0; j < 8; ++j)
            s_tmp[sb + j * 16] = acc[j];
    }
    __syncthreads();

    // ---- pass 2 (vertical): D(16x16) = Wv * tmp ----------------------------
    v2f bv[CHK];
    const int bbase = q0base * 16 + m;
    #pragma unroll
    for (int t = 0; t < CHK; ++t) {
        bv[t].x = s_tmp[bbase + 64 * t];
        bv[t].y = s_tmp[bbase + 64 * t + 16];
    }
    v8f vacc = {0.f, 0.f, 0.f, 0.f, 0.f, 0.f, 0.f, 0.f};
    #pragma unroll
    for (int t = 0; t < CHK; ++t)
        vacc = __builtin_amdgcn_wmma_f32_16x16x4_f32(
            false, wc[t], false, bv[t], (short)0, vacc, false, false);

    #pragma unroll
    for (int j = 0; j < 8; ++j)
        op[(size_t)(half * 8 + j) * WDIM + m] = vacc[j];
}

// ---- one wave = one 16x16 tile of one (plane, blur) ------------------------
__global__ void __launch_bounds__(32)
gauss_sep_wmma(const float* __restrict__ x, float* __restrict__ out)
{
    __shared__ float s_x[TIW * TIW + 4];  // +4: p=9 tail chunk reads 1 past end
    __shared__ float s_tmp[48 * 16];
    __shared__ float s_gpad[64];          // s_gpad[16+d] = g[d] for 0<=d<k, else 0

    const int lane = threadIdx.x;
    const int half = lane >> 4;
    const int m    = lane & 15;

    const int i  = blockIdx.z % NK;
    const int pl = blockIdx.z / NK;       // b*3 + ch
    const int rbase = blockIdx.y << 4;
    const int cbase = blockIdx.x << 4;

    const int k = ksize_of(i);
    const int p = (k - 1) >> 1;

    const float* xp = x + (size_t)pl * HDIM * WDIM;

    // ---- taps: 1 expf/lane + wave32 xor-reduction; zero-padded table -------
    {
        const float sigma  = 0.5f + 0.025f * (float)i;
        const float inv2s2 = 1.0f / (2.0f * sigma * sigma);
        const float d = (float)lane - (float)(k >> 1);
        float gv = (lane < k) ? __expf(-d * d * inv2s2) : 0.0f;
        float s = gv;
        #pragma unroll
        for (int off = 16; off; off >>= 1) s += __shfl_xor(s, off, 32);
        s_gpad[16 + lane] = gv / s;                       // d in [0,32)
        s_gpad[(lane < 16) ? lane : (32 + lane)] = 0.0f;  // zeros both sides
    }

    // ---- stage fixed 36x36 tile (origin rbase-10, cbase-10) into LDS -------
    const int r0 = rbase - 10, c0 = cbase - 10;
    if (r0 >= 0 && c0 >= 0 && r0 + TIW <= HDIM && c0 + TIW <= WDIM) {
        // interior: async global->LDS b64 (ASYNCcnt), coalesced 144B rows
        const float* gbase = xp + (size_t)r0 * WDIM + c0;
        const unsigned lbase = (unsigned)(size_t)(&s_x[0]);
        #pragma unroll 1
        for (int it = 0; it < 20; ++it) {          // 640 of 648 chunks, no guard
            int idx = it * 32 + lane;
            int rr = idx / (TIW / 2);
            int cc = (idx - rr * (TIW / 2)) * 2;
            const float* gp = gbase + rr * WDIM + cc;
            unsigned la = lbase + (unsigned)((rr * TIW + cc) * 4);
            asm volatile("global_load_async_to_lds_b64 %0, %1, off"
                         :: "v"(la), "v"(gp) : "memory");
        }
        if (lane < 8) {                            // tail: chunks 640..647
            int idx = 640 + lane;
            int rr = idx / (TIW / 2);
            int cc = (idx - rr * (TIW / 2)) * 2;
            const float* gp = gbase + rr * WDIM + cc;
            unsigned la = lbase + (unsigned)((rr * TIW + cc) * 4);
            asm volatile("global_load_async_to_lds_b64 %0, %1, off"
                         :: "v"(la), "v"(gp) : "memory");
        }
        asm volatile("s_wait_asynccnt 0x0" ::: "memory");
    } else {
        // border tiles (~12%): reflect per element
        #pragma unroll 1
        for (int it = 0; it < 40; ++it) {          // 1280 of 1296, no guard
            int idx = it * 32 + lane;
            int rr = idx / TIW;
            int cc = idx - rr * TIW;
            s_x[idx] = xp[(size_t)refl(r0 + rr, HDIM) * WDIM + refl(c0 + cc, WDIM)];
        }
        if (lane < 16) {
            int idx = 1280 + lane;
            int rr = idx / TIW;
            int cc = idx - rr * TIW;
            s_x[idx] = xp[(size_t)refl(r0 + rr, HDIM) * WDIM + refl(c0 + cc, WDIM)];
        }
    }
    __syncthreads();

    const int b  = pl / CHN;
    const int ch = pl - b * CHN;
    float* op = out + ((size_t)(b * OCH + CHN + i * CHN + ch) * HDIM + rbase) * WDIM
              + cbase;

    switch (p) {                                   // uniform: one scalar jump
        case 1:  blur_passes<1>(s_x, s_tmp, s_gpad, op, half, m); break;
        case 2:  blur_passes<2>(s_x, s_tmp, s_gpad, op, half, m); break;
        case 3:  blur_passes<3>(s_x, s_tmp, s_gpad, op, half, m); break;
        case 4:  blur_passes<4>(s_x, s_tmp, s_gpad, op, half, m); break;
        case 5:  blur_passes<5>(s_x, s_tmp, s_gpad, op, half, m); break;
        case 6:  blur_passes<6>(s_x, s_tmp, s_gpad, op, half, m); break;
        case 7:  blur_passes<7>(s_x, s_tmp, s_gpad, op, half, m); break;
        case 8:  blur_passes<8>(s_x, s_tmp, s_gpad, op, half, m); break;
        case 9:  blur_passes<9>(s_x, s_tmp, s_gpad, op, half, m); break;
        default: blur_passes<10>(s_x, s_tmp, s_gpad, op, half, m); break;
    }
}

extern "C" void kernel_launch(void* const* d_in, const int* in_sizes, int n_in,
                              void* d_out, int out_size, void* d_ws, size_t ws_size,
                              hipStream_t stream) {
    (void)in_sizes; (void)n_in; (void)out_size; (void)d_ws; (void)ws_size;
    const float* x = (const float*)d_in[0];
    float* out     = (float*)d_out;

    const size_t n4 = (size_t)BATCH * CHN * HDIM * WDIM / 4;
    identity_copy<<<dim3((unsigned)((n4 + 255) / 256)), 256, 0, stream>>>(
        (const float4*)x, (float4*)out);

    dim3 grid(WDIM / 16, HDIM / 16, BATCH * CHN * NK);
    gauss_sep_wmma<<<grid, 32, 0, stream>>>(x, out);
}